// Sampler_86990267613418
// MI455X (gfx1250) — compile-verified
//
#include <hip/hip_runtime.h>
#include <stdint.h>

// Problem constants (from reference)
constexpr int kN = 128;
constexpr int kV = 128000;
constexpr int kPLen = 1024;
constexpr int kOLen = 256;
constexpr int kChunk = 4000;           // 16 KB tile, 32 chunks per row
constexpr int kNChunks = kV / kChunk;  // 32
constexpr int kChunksPerBlk = 4;       // penalty kernel: chunks per block

typedef uint32_t u32x4 __attribute__((ext_vector_type(4)));
typedef int      i32x4 __attribute__((ext_vector_type(4)));
typedef int      i32x8 __attribute__((ext_vector_type(8)));

#if defined(__HIP_DEVICE_COMPILE__) && __has_builtin(__builtin_amdgcn_tensor_load_to_lds)
#define HAVE_TDM 1
#else
#define HAVE_TDM 0
#endif

// Order-preserving float<->uint mapping (larger float -> larger uint)
__device__ __forceinline__ uint32_t f2o(float f) {
  uint32_t u = __float_as_uint(f);
  return (u & 0x80000000u) ? ~u : (u | 0x80000000u);
}
__device__ __forceinline__ float o2f(uint32_t u) {
  uint32_t b = (u & 0x80000000u) ? (u & 0x7FFFFFFFu) : ~u;
  return __uint_as_float(b);
}

#if HAVE_TDM
// Issue a 1-D TDM transfer of n 4-byte elements global -> LDS (per-wave op;
// EXEC is ignored by tensor instructions, so call from one wave only).
__device__ __forceinline__ void tdm_load_1d(const void* g, void* l, int n) {
  uint64_t ga = (uint64_t)(uintptr_t)g;
  uint32_t la = (uint32_t)(uintptr_t)l;  // low 32 bits of flat = LDS offset
  u32x4 g0;
  g0[0] = 1u;                                            // count=1 (valid D#)
  g0[1] = la;                                            // lds_addr
  g0[2] = (uint32_t)ga;                                  // global_addr lo
  g0[3] = (uint32_t)((ga >> 32) & 0x01FFFFFFu) | (2u << 30);  // hi + type=2
  i32x8 g1;
  g1[0] = (int)(2u << 16);                               // data_size = 4 B
  g1[1] = (int)(((uint32_t)n & 0xFFFFu) << 16);          // tensor_dim0 lo16
  g1[2] = (int)(((uint32_t)n >> 16) | (1u << 16));       // dim0 hi16, dim1=1
  g1[3] = (int)((uint32_t)n << 16);                      // tile_dim0 = n
  g1[4] = 1;                                             // tile_dim1=1
  g1[5] = n;                                             // tensor_dim0_stride
  g1[6] = 0;
  g1[7] = 0;
  i32x4 z4 = {0, 0, 0, 0};
#if __clang_major__ >= 23
  i32x8 z8 = {0, 0, 0, 0, 0, 0, 0, 0};
  __builtin_amdgcn_tensor_load_to_lds(g0, g1, z4, z4, z8, 0);
#else
  __builtin_amdgcn_tensor_load_to_lds(g0, g1, z4, z4, 0);
#endif
}
#endif

// ---------------------------------------------------------------------------
// Double-buffered TDM streaming pass over one row of V floats.
// f(i, l) is called once per element with its row index and value.
// ---------------------------------------------------------------------------
template <typename F>
__device__ __forceinline__ void row_pass(const float* __restrict__ row,
                                         float (&tbuf)[2][kChunk], F f) {
#if HAVE_TDM
  if (threadIdx.x < 32) tdm_load_1d(row, tbuf[0], kChunk);
  for (int c = 0; c < kNChunks; ++c) {
    const int b = c & 1;
    if (threadIdx.x < 32) {
      if (c + 1 < kNChunks) {
        tdm_load_1d(row + (size_t)(c + 1) * kChunk, tbuf[b ^ 1], kChunk);
        __builtin_amdgcn_s_wait_tensorcnt(1);   // oldest stage landed
      } else {
        __builtin_amdgcn_s_wait_tensorcnt(0);
      }
    }
    __syncthreads();                            // stage c visible to all waves
    const int base = c * kChunk;
    for (int i = threadIdx.x; i < kChunk; i += blockDim.x)
      f(base + i, tbuf[b][i]);
    __syncthreads();                            // done reading before refill
  }
#else
  for (int i = threadIdx.x; i < kV; i += blockDim.x) f(i, row[i]);
  __syncthreads();
#endif
}

// ---------------------------------------------------------------------------
// K0: zero workspace (token table + per-row max + per-row threshold)
// ---------------------------------------------------------------------------
__global__ void k_init(uint32_t* ws) {
  size_t total = (size_t)kN * kV + 2 * (size_t)kN;
  for (size_t i = (size_t)blockIdx.x * blockDim.x + threadIdx.x; i < total;
       i += (size_t)gridDim.x * blockDim.x)
    ws[i] = 0u;
}

// ---------------------------------------------------------------------------
// K1: scatter token occurrences.
//   tok[r*V+t]: bit31 = token appears in prompt; bits[30:0] = output count
// ---------------------------------------------------------------------------
__global__ void k_scatter(const int* __restrict__ prompt,
                          const int* __restrict__ outtok,
                          uint32_t* __restrict__ tok) {
  int i = blockIdx.x * blockDim.x + threadIdx.x;
  if (i < kN * kPLen) {
    int r = i / kPLen;
    int t = prompt[i];
    atomicOr(&tok[(size_t)r * kV + t], 0x80000000u);
  } else if (i < kN * (kPLen + kOLen)) {
    int j = i - kN * kPLen;
    int r = j / kOLen;
    int t = outtok[j];
    atomicAdd(&tok[(size_t)r * kV + t], 1u);
  }
}

// ---------------------------------------------------------------------------
// K2: apply penalties, write penalized logits, reduce per-row max.
//     Logits AND token-table tiles are staged by a double-buffered TDM
//     pipeline (2 descriptors in flight per stage).
// ---------------------------------------------------------------------------
__global__ __launch_bounds__(256)
void k_penalize(const float* __restrict__ logits,
                const float* __restrict__ pres,
                const float* __restrict__ freq,
                const float* __restrict__ rep,
                const uint32_t* __restrict__ tok,
                float* __restrict__ out,
                uint32_t* __restrict__ rowmax) {
  __shared__ float ltile[2][kChunk];     // 32 KB
  __shared__ uint32_t ttile[2][kChunk];  // 32 KB
  __shared__ uint32_t blkmax;
  const int r = blockIdx.y;
  const int c0 = blockIdx.x * kChunksPerBlk;
  const size_t rowbase = (size_t)r * kV;

  if (threadIdx.x == 0) blkmax = 0u;
  const float fr = freq[r], pr = pres[r], rp = rep[r];
  float lmax = -__builtin_inff();

#if HAVE_TDM
  if (threadIdx.x < 32) {
    const size_t b0 = rowbase + (size_t)c0 * kChunk;
    tdm_load_1d(logits + b0, ltile[0], kChunk);
    tdm_load_1d(tok + b0, ttile[0], kChunk);
  }
  for (int cc = 0; cc < kChunksPerBlk; ++cc) {
    const int b = cc & 1;
    const size_t base = rowbase + (size_t)(c0 + cc) * kChunk;
    if (threadIdx.x < 32) {
      if (cc + 1 < kChunksPerBlk) {
        const size_t nbase = rowbase + (size_t)(c0 + cc + 1) * kChunk;
        tdm_load_1d(logits + nbase, ltile[b ^ 1], kChunk);
        tdm_load_1d(tok + nbase, ttile[b ^ 1], kChunk);
        __builtin_amdgcn_s_wait_tensorcnt(2);  // older stage (2 ops) landed
      } else {
        __builtin_amdgcn_s_wait_tensorcnt(0);
      }
    }
    __syncthreads();
    for (int i = threadIdx.x; i < kChunk; i += blockDim.x) {
      float l = ltile[b][i];
      uint32_t tv = ttile[b][i];
      uint32_t cnt = tv & 0x7FFFFFFFu;
      if (tv != 0u) l = (l > 0.f) ? (l / rp) : (l * rp);  // repetition
      l -= fr * (float)cnt;                               // frequency
      if (cnt != 0u) l -= pr;                             // presence
      out[base + i] = l;
      lmax = fmaxf(lmax, l);
    }
    __syncthreads();
  }
#else
  __syncthreads();
  for (int cc = 0; cc < kChunksPerBlk; ++cc) {
    const size_t base = rowbase + (size_t)(c0 + cc) * kChunk;
    for (int i = threadIdx.x; i < kChunk; i += blockDim.x) {
      float l = logits[base + i];
      uint32_t tv = tok[base + i];
      uint32_t cnt = tv & 0x7FFFFFFFu;
      if (tv != 0u) l = (l > 0.f) ? (l / rp) : (l * rp);
      l -= fr * (float)cnt;
      if (cnt != 0u) l -= pr;
      out[base + i] = l;
      lmax = fmaxf(lmax, l);
    }
  }
  __syncthreads();
#endif
  atomicMax(&blkmax, f2o(lmax));
  __syncthreads();
  if (threadIdx.x == 0) atomicMax(&rowmax[r], blkmax);
}

// ---------------------------------------------------------------------------
// K3: per-row threshold search (3-level 1024-way radix histogram) + apply.
//   top-k  -> threshold t_k (k-th order statistic, 30-bit refined)
//   top-p  -> threshold t_p (exp-weighted histogram scan)
//   min-p  -> algebraic cut  l >= max + ln(min_p)   (no second softmax!)
//   keep iff  f2o(l) >= max(t_k, t_p)  &&  l >= cut
// All four row passes stream through the double-buffered TDM pipeline.
// ---------------------------------------------------------------------------
__global__ __launch_bounds__(1024)
void k_thresh_apply(float* __restrict__ out,
                    const uint32_t* __restrict__ rowmax,
                    const float* __restrict__ pv,
                    const float* __restrict__ minpv,
                    const int* __restrict__ kv,
                    uint32_t* __restrict__ thr) {
  const int r = blockIdx.x;
  float* row = out + (size_t)r * kV;
  const float maxv = o2f(rowmax[r]);
  const float p = pv[r];
  int k = kv[r];
  if (k < 1) k = 1;
  if (k > kV) k = kV;

  __shared__ float tbuf[2][kChunk];  // 32 KB TDM staging
  __shared__ uint32_t hcnt[1024];    // 4 KB
  __shared__ float hsum[1024];       // 4 KB
  __shared__ uint32_t s_kbase, s_krem, s_pbase;
  __shared__ float s_prem;
  __shared__ int s_pdone;

  // ---- Level 0: full-range histogram (counts + exp sums) ----
  hcnt[threadIdx.x] = 0u;
  hsum[threadIdx.x] = 0.f;
  __syncthreads();
  row_pass(row, tbuf, [&](int i, float l) {
    (void)i;
    uint32_t u = f2o(l);
    atomicAdd(&hcnt[u >> 22], 1u);
    atomicAdd(&hsum[u >> 22], __expf(l - maxv));  // ds_add_f32 in LDS
  });
  if (threadIdx.x == 0) {
    float Z = 0.f;
    for (int b = 0; b < 1024; ++b) Z += hsum[b];
    float pZ = p * Z;
    uint32_t acc = 0;  // k scan (descending)
    for (int b = 1023; b >= 0; --b) {
      if (acc + hcnt[b] >= (uint32_t)k) {
        s_kbase = (uint32_t)b << 22;
        s_krem = (uint32_t)k - acc;
        break;
      }
      acc += hcnt[b];
    }
    float facc = 0.f;  // p scan (descending exclusive prob-mass prefix)
    int pb = -1;
    float prem = 0.f;
    for (int b = 1023; b >= 0; --b) {
      if (facc + hsum[b] > pZ) { pb = b; prem = pZ - facc; break; }
      facc += hsum[b];
    }
    if (pb < 0) { s_pbase = 0u; s_pdone = 1; }
    else        { s_pbase = (uint32_t)pb << 22; s_prem = prem; s_pdone = 0; }
  }
  __syncthreads();

  // ---- Levels 1 and 2: refine inside boundary buckets ----
  for (int shift = 12; shift >= 2; shift -= 10) {
    hcnt[threadIdx.x] = 0u;
    hsum[threadIdx.x] = 0.f;
    __syncthreads();
    const uint32_t kpref = s_kbase >> (shift + 10);
    const uint32_t ppref = s_pbase >> (shift + 10);
    const int pdone = s_pdone;
    row_pass(row, tbuf, [&](int i, float l) {
      (void)i;
      uint32_t u = f2o(l);
      if ((u >> (shift + 10)) == kpref)
        atomicAdd(&hcnt[(u >> shift) & 1023u], 1u);
      if (!pdone && (u >> (shift + 10)) == ppref)
        atomicAdd(&hsum[(u >> shift) & 1023u], __expf(l - maxv));
    });
    if (threadIdx.x == 0) {
      uint32_t acc = 0, krem = s_krem;
      for (int b = 1023; b >= 0; --b) {
        if (acc + hcnt[b] >= krem) {
          s_kbase |= (uint32_t)b << shift;
          s_krem = krem - acc;
          break;
        }
        acc += hcnt[b];
      }
      if (!s_pdone) {
        float facc = 0.f, prem = s_prem;
        int found = 0;
        for (int b = 1023; b >= 0; --b) {
          if (facc + hsum[b] > prem) {
            s_pbase |= (uint32_t)b << shift;
            s_prem = prem - facc;
            found = 1;
            break;
          }
          facc += hsum[b];
        }
        if (!found) s_pdone = 1;  // float-atomic drift: keep whole bucket
      }
    }
    __syncthreads();
  }

  // ---- Apply combined thresholds (streamed via TDM as well) ----
  const uint32_t t = (s_kbase > s_pbase) ? s_kbase : s_pbase;
  if (threadIdx.x == 0) thr[r] = t;
  const float cut = maxv + __logf(minpv[r]);
  row_pass(row, tbuf, [&](int i, float l) {
    bool keep = (f2o(l) >= t) && (l >= cut);
    row[i] = keep ? l : -__builtin_inff();
  });
}

// ---------------------------------------------------------------------------
extern "C" void kernel_launch(void* const* d_in, const int* in_sizes, int n_in,
                              void* d_out, int out_size, void* d_ws, size_t ws_size,
                              hipStream_t stream) {
  (void)in_sizes; (void)n_in; (void)out_size; (void)ws_size;
  const float* logits     = (const float*)d_in[0];
  const float* presence   = (const float*)d_in[1];
  const float* frequency  = (const float*)d_in[2];
  const float* repetition = (const float*)d_in[3];
  const float* p          = (const float*)d_in[4];
  const float* min_p      = (const float*)d_in[5];
  const int*   prompt     = (const int*)d_in[6];
  const int*   outtok     = (const int*)d_in[7];
  const int*   kk         = (const int*)d_in[8];
  float* out = (float*)d_out;

  uint32_t* tok    = (uint32_t*)d_ws;          // N*V u32 token table
  uint32_t* rowmax = tok + (size_t)kN * kV;    // N u32 (ordered-float max)
  uint32_t* thr    = rowmax + kN;              // N u32 thresholds

  k_init<<<8192, 256, 0, stream>>>(tok);
  int scat = kN * (kPLen + kOLen);
  k_scatter<<<(scat + 255) / 256, 256, 0, stream>>>(prompt, outtok, tok);
  dim3 pg(kV / (kChunk * kChunksPerBlk), kN);  // (8, 128)
  k_penalize<<<pg, 256, 0, stream>>>(logits, presence, frequency, repetition,
                                     tok, out, rowmax);
  k_thresh_apply<<<kN, 1024, 0, stream>>>(out, rowmax, p, min_p, kk, thr);
}